// ROIPooler_10831907520636
// MI455X (gfx1250) — compile-verified
//
#include <hip/hip_runtime.h>
#include <hip/hip_bf16.h>

typedef float v2f __attribute__((ext_vector_type(2)));
typedef float v8f __attribute__((ext_vector_type(8)));

#define PP      7      // output bins per axis
#define SAMP    14     // P*S sample points per axis
#define NSAMP   196    // 14*14
#define CCH     64     // channels handled per block
#define CTOT    256    // total channels
#define NTHREADS 256

__global__ __launch_bounds__(NTHREADS)
void roi_align_wmma_kernel(const float* __restrict__ f2,
                           const float* __restrict__ f3,
                           const float* __restrict__ f4,
                           const float* __restrict__ f5,
                           const float* __restrict__ boxes,
                           float* __restrict__ out)
{
    // LDS: sample matrix V[sample][channel], padded stride 65 (bank-conflict free
    // for sample-major writes and channel-contiguous WMMA reads). ~51 KB total.
    __shared__ float Vt[NSAMP][CCH + 1];
    __shared__ int   s_bin[NSAMP];
    __shared__ int   s_ylo[SAMP], s_yhi[SAMP], s_xlo[SAMP], s_xhi[SAMP];
    __shared__ float s_yfr[SAMP], s_xfr[SAMP];
    __shared__ int   s_yv[SAMP],  s_xv[SAMP];

    const int tid = threadIdx.x;
    const int roi = blockIdx.x;        // 0..1023  (N*K)
    const int c0  = blockIdx.y * CCH;  // channel chunk base

    // ---- per-ROI scalars (uniform across block; computed redundantly) ----
    const float bx1 = boxes[roi * 4 + 0];
    const float by1 = boxes[roi * 4 + 1];
    const float bx2 = boxes[roi * 4 + 2];
    const float by2 = boxes[roi * 4 + 3];

    const float rsize = sqrtf((bx2 - bx1) * (by2 - by1));
    float lv = floorf(4.0f + log2f(rsize * (1.0f / 224.0f) + 1e-8f));
    lv = fminf(fmaxf(lv, 2.0f), 5.0f);
    const int   l     = (int)lv - 2;               // 0..3
    const float scale = 0.25f / (float)(1 << l);   // 0.25 .. 0.03125
    const int   H     = 256 >> l;
    const int   W     = H;
    const float* feat = (l == 0) ? f2 : (l == 1) ? f3 : (l == 2) ? f4 : f5;
    const int   n     = roi >> 9;                  // batch index (K = 512)

    // box -> feature coords (aligned=True: -0.5)
    const float fx1 = bx1 * scale - 0.5f, fy1 = by1 * scale - 0.5f;
    const float fx2 = bx2 * scale - 0.5f, fy2 = by2 * scale - 0.5f;
    const float bin_h = (fy2 - fy1) * (1.0f / 7.0f);
    const float bin_w = (fx2 - fx1) * (1.0f / 7.0f);

    // ---- per-axis sample prep (14 y + 14 x) ----
    if (tid < SAMP) {
        const float g = ((float)tid + 0.5f) * 0.5f;
        const float y = fy1 + g * bin_h;
        const int   v = (y >= -1.0f) && (y <= (float)H);
        const float yc = fmaxf(y, 0.0f);
        int lo = (int)floorf(yc); if (lo > H - 1) lo = H - 1;
        int hi = lo + 1;          if (hi > H - 1) hi = H - 1;
        s_ylo[tid] = lo; s_yhi[tid] = hi;
        s_yfr[tid] = yc - (float)lo; s_yv[tid] = v;
    } else if (tid < 2 * SAMP) {
        const int   j = tid - SAMP;
        const float g = ((float)j + 0.5f) * 0.5f;
        const float x = fx1 + g * bin_w;
        const int   v = (x >= -1.0f) && (x <= (float)W);
        const float xc = fmaxf(x, 0.0f);
        int lo = (int)floorf(xc); if (lo > W - 1) lo = W - 1;
        int hi = lo + 1;          if (hi > W - 1) hi = W - 1;
        s_xlo[j] = lo; s_xhi[j] = hi;
        s_xfr[j] = xc - (float)lo; s_xv[j] = v;
    }
    if (tid < NSAMP) {               // sample -> output-bin table (constant matrix)
        const int i = tid / SAMP;
        const int j = tid - i * SAMP;
        s_bin[tid] = (i >> 1) * PP + (j >> 1);
    }
    __syncthreads();

    // ---- gather stage: build V[196][64] (bilinear samples) ----
    // sample-major across lanes => adjacent lanes touch adjacent x in the same
    // feature row (cacheline coherent; the whole pyramid fits the 192MB L2).
    const int HW = H * W;
    const float* fb = feat + (size_t)n * CTOT * HW;
#pragma unroll 1
    for (int it = 0; it < (NSAMP * CCH) / NTHREADS; ++it) {   // 49 iterations
        const int idx = it * NTHREADS + tid;
        const int c = idx / NSAMP;
        const int s = idx - c * NSAMP;
        const int i = s / SAMP;
        const int j = s - i * SAMP;
        const float* plane = fb + (size_t)(c0 + c) * HW;
        const int yl = s_ylo[i], yh = s_yhi[i];
        const int xl = s_xlo[j], xh = s_xhi[j];
        const float wy = s_yfr[i], wx = s_xfr[j];
        const float f00 = plane[yl * W + xl];
        const float f01 = plane[yl * W + xh];
        const float f10 = plane[yh * W + xl];
        const float f11 = plane[yh * W + xh];
        const float top = f00 + wx * (f01 - f00);
        const float bot = f10 + wx * (f11 - f10);
        float v = top + wy * (bot - top);
        if (!(s_yv[i] & s_xv[j])) v = 0.0f;
        Vt[s][c] = v;
    }
    __syncthreads();

    // ---- WMMA stage: OutT[ch, pos] = V^T (16xK) x Wbin (Kx16), K in steps of 4 ----
    const int lane = tid & 31;
    const int half = lane >> 4;       // 0: lanes 0-15, 1: lanes 16-31
    const int ln   = lane & 15;
    // Wave id forced into an SGPR: makes the tile index and the K-loop bounds
    // provably wave-uniform, so both loops compile to scalar branches and EXEC
    // stays all-ones across every WMMA (required by the ISA).
    const int wv_s = __builtin_amdgcn_readfirstlane(tid >> 5);   // 8 waves

    for (int t = wv_s; t < 16; t += 8) {        // 4 M-tiles x 4 N-tiles (scalar)
        const int mt = t >> 2;                  // channel tile (16 ch)
        const int nt = t & 3;                   // position tile (16 pos)
        const int crow = mt * 16 + ln;          // A-matrix row = channel
        const int pcol = nt * 16 + ln;          // B/D column = output position

        // Scalar K-range whose bins can hit this position tile (rest is zero B)
        const int pylo = (16 * nt) / PP;
        int phi = 16 * nt + 15; if (phi > 48) phi = 48;
        const int pyhi = phi / PP;
        const int kb_lo = 28 * pylo;            // multiples of 4
        const int kb_hi = 28 * (pyhi + 1);

        v8f acc = {0.f, 0.f, 0.f, 0.f, 0.f, 0.f, 0.f, 0.f};
#pragma unroll 1
        for (int kb = kb_lo; kb < kb_hi; kb += 4) {
            const int k0 = kb + (half << 1);    // VGPR r holds K = k0 + r
            v2f a, b;
            a.x = Vt[k0][crow];                 // A = V^T tile (16x4 f32)
            a.y = Vt[k0 + 1][crow];
            b.x = (s_bin[k0]     == pcol) ? 0.25f : 0.0f;   // B = bin weights (4x16)
            b.y = (s_bin[k0 + 1] == pcol) ? 0.25f : 0.0f;
            acc = __builtin_amdgcn_wmma_f32_16x16x4_f32(
                      false, a, false, b, (short)0, acc, false, false);
        }

        // D: N = pcol, VGPR r -> channel = mt*16 + r + 8*half. Stores are
        // contiguous across lanes (stride-1 in p) => coalesced.
        if (pcol < PP * PP) {
            const size_t obase =
                ((size_t)roi * CTOT + (size_t)(c0 + mt * 16 + 8 * half)) * (PP * PP)
                + (size_t)pcol;
#pragma unroll
            for (int r = 0; r < 8; ++r)
                out[obase + (size_t)r * (PP * PP)] = acc[r];
        }
    }
}

extern "C" void kernel_launch(void* const* d_in, const int* in_sizes, int n_in,
                              void* d_out, int out_size, void* d_ws, size_t ws_size,
                              hipStream_t stream) {
    const float* f2    = (const float*)d_in[0];
    const float* f3    = (const float*)d_in[1];
    const float* f4    = (const float*)d_in[2];
    const float* f5    = (const float*)d_in[3];
    const float* boxes = (const float*)d_in[4];
    float* out = (float*)d_out;

    const int n_rois = in_sizes[4] / 4;          // N*K = 1024
    dim3 grid(n_rois, CTOT / CCH);               // 1024 x 4 channel chunks
    roi_align_wmma_kernel<<<grid, NTHREADS, 0, stream>>>(f2, f3, f4, f5, boxes, out);
}